// GraphSageConv_76192719831688
// MI455X (gfx1250) — compile-verified
//
#include <hip/hip_runtime.h>

typedef __attribute__((ext_vector_type(2))) float v2f;
typedef __attribute__((ext_vector_type(8))) float v8f;

#define NN 100000
#define DIM 128
#define NE 1600000

// ---------------- zero fill (float4 stores) ----------------
__global__ __launch_bounds__(256) void zero_f32(float* __restrict__ p, int n4) {
  int t = blockIdx.x * 256 + threadIdx.x;
  if (t < n4) ((float4*)p)[t] = make_float4(0.f, 0.f, 0.f, 0.f);
}

// ---------------- he_aggr = segment_sum(edge_feat, dst) ----------------
__global__ __launch_bounds__(256) void scatter_edge_feat(const float* __restrict__ ef,
                                                         const int* __restrict__ dst,
                                                         float* __restrict__ aggr) {
  int t = blockIdx.x * 256 + threadIdx.x;
  if (t < NE) unsafeAtomicAdd(&aggr[dst[t]], ef[t]);
}

// ---------------- hn_aggr = segment_sum(h[src], dst) ----------------
// one wave32 per edge: 32 lanes x float4 = the whole 128-float row (512B coalesced)
__global__ __launch_bounds__(256) void scatter_node_feat(const float* __restrict__ h,
                                                         const int* __restrict__ src,
                                                         const int* __restrict__ dst,
                                                         float* __restrict__ aggr) {
  int w = (blockIdx.x * 256 + threadIdx.x) >> 5;
  int lane = threadIdx.x & 31;
  if (w >= NE) return;
  int s = src[w];
  int d = dst[w];
  float4 v = ((const float4*)(h + (size_t)s * DIM))[lane];
  float* ap = aggr + (size_t)d * DIM + lane * 4;
  unsafeAtomicAdd(ap + 0, v.x);
  unsafeAtomicAdd(ap + 1, v.y);
  unsafeAtomicAdd(ap + 2, v.z);
  unsafeAtomicAdd(ap + 3, v.w);
}

// ---------------- fused SAGE dense layer via V_WMMA_F32_16X16X4_F32 ----------------
// out[n, c] = relu( sum_k concat(hA[n], hB[n], he[n])[k] * W[c, k] + bias[c] )
// One wave computes a 16(M=nodes) x 16(N=out feats) tile, K stepped by 4.
// A-frag layout (32-bit 16x4): lane L -> row L&15, K = k0 + 2*(L>>4) + {0,1}
// B-frag layout (32-bit 4x16): lane L -> col L&15, K = k0 + 2*(L>>4) + {0,1}
// C/D layout: VGPR i, lanes 0-15: M=i ; lanes 16-31: M=i+8 ; N = L&15
template <bool HAS_AGGR, bool RELU>
__global__ __launch_bounds__(256) void sage_gemm(const float* __restrict__ hA,
                                                 const float* __restrict__ hB,
                                                 const float* __restrict__ he,
                                                 const float* __restrict__ W,
                                                 const float* __restrict__ bias,
                                                 float* __restrict__ out, int ldw) {
  const int tm = blockIdx.x;         // 0..6249 (node tiles)
  const int tn = threadIdx.x >> 5;   // 0..7    (feature tiles, one per wave)
  const int lane = threadIdx.x & 31;
  const int m_base = tm * 16;
  const int n_base = tn * 16;
  const int r = lane & 15;           // A row / B-C-D column-in-tile
  const int hi = lane >> 4;          // K-half select
  const int col = n_base + r;        // output feature index

  const float* arow = hA + (size_t)(m_base + r) * DIM;
  const float* wrow = W + (size_t)col * ldw;

  v8f acc = {};
#pragma unroll 4
  for (int k0 = 0; k0 < DIM; k0 += 4) {
    int ka = k0 + 2 * hi;
    v2f a = *(const v2f*)(arow + ka);
    v2f b;
    b[0] = wrow[ka];
    b[1] = wrow[ka + 1];
    acc = __builtin_amdgcn_wmma_f32_16x16x4_f32(false, a, false, b, (short)0, acc,
                                                false, false);
  }
  if (HAS_AGGR) {
    const float* brow = hB + (size_t)(m_base + r) * DIM;
    const float* wrow2 = wrow + DIM;
#pragma unroll 4
    for (int k0 = 0; k0 < DIM; k0 += 4) {
      int ka = k0 + 2 * hi;
      v2f a = *(const v2f*)(brow + ka);
      v2f b;
      b[0] = wrow2[ka];
      b[1] = wrow2[ka + 1];
      acc = __builtin_amdgcn_wmma_f32_16x16x4_f32(false, a, false, b, (short)0, acc,
                                                  false, false);
    }
  }

  const float bv = bias[col];
  const float wlast = HAS_AGGR ? wrow[2 * DIM] : 0.0f;  // k = 256 (he_aggr column)
#pragma unroll
  for (int i = 0; i < 8; ++i) {
    int node = m_base + i + 8 * hi;
    float v = acc[i];
    if (HAS_AGGR) v += he[node] * wlast;
    v += bv;
    if (RELU) v = fmaxf(v, 0.0f);
    out[(size_t)node * DIM + col] = v;
  }
}

// ---------------- final head: out[n] = h[n,:] . w + b ----------------
__global__ __launch_bounds__(256) void final_matvec(const float* __restrict__ h,
                                                    const float* __restrict__ w,
                                                    const float* __restrict__ b,
                                                    float* __restrict__ out) {
  int node = (blockIdx.x * 256 + threadIdx.x) >> 5;
  int lane = threadIdx.x & 31;
  if (node >= NN) return;
  float4 hv = ((const float4*)(h + (size_t)node * DIM))[lane];
  float4 wv = ((const float4*)w)[lane];
  float s = hv.x * wv.x + hv.y * wv.y + hv.z * wv.z + hv.w * wv.w;
#pragma unroll
  for (int off = 16; off > 0; off >>= 1) s += __shfl_down(s, off, 32);
  if (lane == 0) out[node] = s + b[0];
}

extern "C" void kernel_launch(void* const* d_in, const int* in_sizes, int n_in,
                              void* d_out, int out_size, void* d_ws, size_t ws_size,
                              hipStream_t stream) {
  (void)in_sizes; (void)n_in; (void)out_size; (void)ws_size;

  const float* node_feat = (const float*)d_in[0];
  const float* edge_feat = (const float*)d_in[1];
  const int* src = (const int*)d_in[2];
  const int* dst = (const int*)d_in[3];
  const float* Wl[5];
  const float* bl[5];
  for (int i = 0; i < 5; ++i) {
    Wl[i] = (const float*)d_in[4 + 2 * i];
    bl[i] = (const float*)d_in[5 + 2 * i];
  }
  const float* Wr1 = (const float*)d_in[14];
  const float* br1 = (const float*)d_in[15];
  const float* Wr2 = (const float*)d_in[16];
  const float* br2 = (const float*)d_in[17];
  const float* Wr3 = (const float*)d_in[18];
  const float* br3 = (const float*)d_in[19];

  float* ws = (float*)d_ws;
  float* he_aggr = ws;                                   // 100000 f32 (+pad)
  float* hn_aggr = ws + 102400;                          // 100000*128 f32
  float* h0 = hn_aggr + (size_t)NN * DIM;                // ping
  float* h1 = h0 + (size_t)NN * DIM;                     // pong

  const int ZB = (NN * DIM / 4) / 256;  // 12500 blocks to zero hn_aggr
  const int SB = NE / 8;                // 200000 blocks, 8 waves each = 1 wave/edge
  const int GB = NN / 16;               // 6250 blocks, 8 waves = full tile grid
  const int MB = NN / 8;                // 12500 blocks for matvec

  // edge-feature aggregation is loop-invariant: compute once
  zero_f32<<<(NN / 4 + 255) / 256, 256, 0, stream>>>(he_aggr, NN / 4);
  scatter_edge_feat<<<NE / 256, 256, 0, stream>>>(edge_feat, dst, he_aggr);

  // 5 SAGE layers
  const float* hin = node_feat;
  float* bufs[2] = {h0, h1};
  for (int l = 0; l < 5; ++l) {
    zero_f32<<<ZB, 256, 0, stream>>>(hn_aggr, NN * DIM / 4);
    scatter_node_feat<<<SB, 256, 0, stream>>>(hin, src, dst, hn_aggr);
    float* hout = bufs[l & 1];
    sage_gemm<true, true><<<GB, 256, 0, stream>>>(hin, hn_aggr, he_aggr, Wl[l],
                                                  bl[l], hout, 2 * DIM + 1);
    hin = hout;
  }

  // regression branch
  zero_f32<<<ZB, 256, 0, stream>>>(hn_aggr, NN * DIM / 4);
  scatter_node_feat<<<SB, 256, 0, stream>>>(hin, src, dst, hn_aggr);
  float* t1 = (hin == h0) ? h1 : h0;
  sage_gemm<true, true><<<GB, 256, 0, stream>>>(hin, hn_aggr, he_aggr, Wr1, br1,
                                                t1, 2 * DIM + 1);
  float* t2 = (t1 == h0) ? h1 : h0;
  sage_gemm<false, true><<<GB, 256, 0, stream>>>(t1, nullptr, nullptr, Wr2, br2,
                                                 t2, DIM);
  final_matvec<<<MB, 256, 0, stream>>>(t2, Wr3, br3, (float*)d_out);
}